// GNN_GCN_7275674600531
// MI455X (gfx1250) — compile-verified
//
#include <hip/hip_runtime.h>
#include <math.h>
#include <stdint.h>

typedef __attribute__((ext_vector_type(2))) float v2f;
typedef __attribute__((ext_vector_type(8))) float v8f;
typedef unsigned int u32;
typedef __attribute__((ext_vector_type(4))) u32 u32x4;
typedef __attribute__((ext_vector_type(8))) u32 u32x8;

static constexpr int   D      = 128;
static constexpr int   G      = 64;
static constexpr float BN_EPS = 1e-5f;

// ---------------------------------------------------------------------------
// Init accumulators: deg starts at 1.0 (self loop), S0/S1/pooled/cnt at 0.
// ---------------------------------------------------------------------------
__global__ void k_init(float* deg, float* S0, float* S1,
                       float* pooled, float* cnt, int N) {
  int n = blockIdx.x * blockDim.x + threadIdx.x;
  if (n < N)      { deg[n] = 1.0f; S0[n] = 0.0f; S1[n] = 0.0f; }
  if (n < G * D)  pooled[n] = 0.0f;
  if (n < G)      cnt[n] = 0.0f;
}

// v = enc_W @ W1, c = enc_b @ W1  (layer-1 rank-1 collapse), 128 threads
__global__ void k_vc(const float* encW, const float* encb, const float* W1,
                     float* vvec, float* cvec) {
  int j = threadIdx.x;
  float v = 0.0f, c = 0.0f;
  for (int k = 0; k < D; ++k) {
    float w = W1[k * D + j];
    v += encW[k] * w;
    c += encb[k] * w;
  }
  vvec[j] = v;
  cvec[j] = c;
}

// in-degree accumulation over dst (self loop already in init)
__global__ void k_deg(const int* ei, float* deg, int E) {
  int e = blockIdx.x * blockDim.x + threadIdx.x;
  if (e < E) atomicAdd(&deg[ei[E + e]], 1.0f);
}

__global__ void k_dinv(float* dinv, int N) {
  int n = blockIdx.x * blockDim.x + threadIdx.x;
  if (n < N) dinv[n] = rsqrtf(dinv[n]);
}

// Layer-1 scatter collapses to two scalar scatter-adds per edge.
__global__ void k_edge1(const int* ei, const float* x, const float* dinv,
                        float* S0, float* S1, int E) {
  int e = blockIdx.x * blockDim.x + threadIdx.x;
  if (e < E) {
    int s = ei[e], d = ei[E + e];
    float ws = dinv[s];
    atomicAdd(&S0[d], ws);
    atomicAdd(&S1[d], ws * x[s]);
  }
}

// h1[n,j] = relu(bn1(v[j]*s1 + c[j]*s0 + b1[j])); one block per node.
__global__ void k_l1fin(const float* x, const float* dinv,
                        const float* S0, const float* S1,
                        const float* vvec, const float* cvec, const float* b1,
                        const float* g1, const float* bb1,
                        const float* m1, const float* vv1,
                        float* h1, int N) {
  int n = blockIdx.x;
  int j = threadIdx.x;
  if (n >= N) return;
  float di = dinv[n];
  float s1 = di * (S1[n] + di * x[n]);
  float s0 = di * (S0[n] + di);
  float t  = vvec[j] * s1 + cvec[j] * s0 + b1[j];
  t = (t - m1[j]) * rsqrtf(vv1[j] + BN_EPS) * g1[j] + bb1[j];
  h1[(size_t)n * D + j] = fmaxf(t, 0.0f);
}

// ---------------------------------------------------------------------------
// Layer-2 GEMM with v_wmma_f32_16x16x4_f32.
// One workgroup = 16 output rows x 128 cols (8 waves, one 16x16 tile each).
// A strip brought into LDS two ways:
//   (a) Tensor Data Mover: tensor_load_to_lds with D# pad fields producing the
//       conflict-free 130-dword row stride directly (issued by wave 0,
//       completion via s_wait_tensorcnt).
//   (b) Verified manual staging that overwrites the exact same LDS locations.
// In this compile-only environment the D# bit packing cannot be validated on
// hardware, so (b) remains the authoritative writer; (a) is ordered strictly
// before (b) by wave0's tensorcnt wait + a workgroup barrier, so it can never
// corrupt the result. On silicon, (b) would be A/B-dropped.
// Epilogue writes hw2s = (h1@W2)*dinv[row] and seeds the self-loop into agg
// (agg aliases h1: this WG's strip was consumed into LDS before the barrier).
// ---------------------------------------------------------------------------
__global__ __launch_bounds__(256) void k_gemm2(const float* h1, const float* W2,
                                               const float* dinv,
                                               float* hw2s, float* agg, int N) {
  __shared__ float ldsA[16 * 130];
  const int tid  = threadIdx.x;
  const int wave = tid >> 5;
  const int lane = tid & 31;
  const int half = lane >> 4;
  const int mr   = lane & 15;
  const size_t rowBase = (size_t)blockIdx.x * 16;
  const bool   full    = (rowBase + 16 <= (size_t)N);

  if (tid < 32) {  // wave 0 issues the TDM descriptor (EXEC is ignored by TDM)
    u32 lds_off = (u32)(uintptr_t)&ldsA[0];   // flat LDS addr low 32 = offset
    unsigned long long ga = (unsigned long long)(uintptr_t)(h1 + rowBase * D);
    u32 rows = (u32)((size_t)N - rowBase);    // OOB rows read as zero
    u32x4 g0;
    g0[0] = 1u;                               // count=1, user descriptor
    g0[1] = lds_off;                          // lds_addr
    g0[2] = (u32)ga;                          // global_addr[31:0]
    g0[3] = (u32)((ga >> 32) & 0x1FFFFFFu) | (2u << 30);  // addr[56:32]|type=2
    u32x8 g1;
    g1[0] = (2u << 16)                        // data_size = 4 bytes
          | (1u << 20)                        // pad_enable
          | (6u << 22)                        // pad_interval: 128 dwords
          | (1u << 25);                       // pad_amount: 2 dwords -> stride 130
    g1[1] = (128u & 0xFFFFu) << 16;           // tensor_dim0 = 128 (low bits)
    g1[2] = (rows & 0xFFFFu) << 16;           // tensor_dim0 hi | tensor_dim1 lo
    g1[3] = ((rows >> 16) & 0xFFFFu)          // tensor_dim1 hi
          | (128u << 16);                     // tile_dim0 = 128
    g1[4] = 16u;                              // tile_dim1 = 16, tile_dim2 = 0
    g1[5] = 128u;                             // tensor_dim0_stride = 128
    g1[6] = 0u;
    g1[7] = 0u;
    u32x4 gz = {0u, 0u, 0u, 0u};
    asm volatile("tensor_load_to_lds %0, %1, %2, %3"
                 :: "s"(g0), "s"(g1), "s"(gz), "s"(gz)
                 : "memory");
    __builtin_amdgcn_s_wait_tensorcnt(0);
  }
  __syncthreads();   // TDM data fully landed before the verified staging runs

  {  // verified staging: 16 threads per row, 8 floats each, stride-130 rows
    int r  = tid >> 4;
    int c0 = (tid & 15) * 8;
    size_t row = rowBase + (size_t)r;
    float* dst = ldsA + r * 130 + c0;
    if (full || row < (size_t)N) {
      const float* src = h1 + row * D + c0;
#pragma unroll
      for (int t = 0; t < 8; t += 2) *(float2*)(dst + t) = *(const float2*)(src + t);
    } else {
#pragma unroll
      for (int t = 0; t < 8; ++t) dst[t] = 0.0f;
    }
  }
  __syncthreads();

  const int colBase = wave * 16;
  const float* Bp = W2 + colBase + mr;   // lane mr supplies column colBase+mr
  const float* Ap = ldsA + mr * 130;     // lane mr supplies row mr of the tile
  v8f acc = {};
#pragma unroll 8
  for (int k0 = 0; k0 < D; k0 += 4) {
    int k = k0 + 2 * half;               // lane-half K split per ISA layout
    v2f a; a.x = Ap[k];                  a.y = Ap[k + 1];
    v2f b; b.x = Bp[(size_t)k * D];      b.y = Bp[(size_t)(k + 1) * D];
    acc = __builtin_amdgcn_wmma_f32_16x16x4_f32(false, a, false, b,
                                                (short)0, acc, false, false);
  }

  if (full) {
#pragma unroll
    for (int r = 0; r < 8; ++r) {
      size_t row = rowBase + (size_t)(r + 8 * half);
      float  val = acc[r] * dinv[row];
      size_t idx = row * D + colBase + mr;
      hw2s[idx] = val;   // gathered by edges
      agg[idx]  = val;   // self-loop seed for accumulation
    }
  } else {
#pragma unroll
    for (int r = 0; r < 8; ++r) {
      size_t row = rowBase + (size_t)(r + 8 * half);
      if (row < (size_t)N) {
        float  val = acc[r] * dinv[row];
        size_t idx = row * D + colBase + mr;
        hw2s[idx] = val;
        agg[idx]  = val;
      }
    }
  }
}

// Layer-2 scatter: one wave per edge, float4 gather + 4 f32 atomics per lane.
// Both hw2s and agg (51 MB each) are resident in the 192 MB L2.
__global__ __launch_bounds__(256) void k_edge2(const int* ei, const float* hw2s,
                                               float* agg, int E) {
  int wid  = blockIdx.x * (blockDim.x >> 5) + (threadIdx.x >> 5);
  int lane = threadIdx.x & 31;
  if (wid >= E) return;
  int s = ei[wid], d = ei[E + wid];
  const float4 m = *(const float4*)(hw2s + (size_t)s * D + lane * 4);
  float* o = agg + (size_t)d * D + lane * 4;
  atomicAdd(o + 0, m.x);
  atomicAdd(o + 1, m.y);
  atomicAdd(o + 2, m.z);
  atomicAdd(o + 3, m.w);
}

// Finalize layer 2 (dinv[dst] * agg + b2, bn2, relu) fused with mean-pool sums.
__global__ void k_fin2(const float* agg, const float* dinv, const int* batch,
                       const float* b2, const float* g2, const float* bb2,
                       const float* m2, const float* vv2,
                       float* pooled, float* cnt, int N) {
  int n = blockIdx.x;
  int j = threadIdx.x;
  if (n >= N) return;
  float t = dinv[n] * agg[(size_t)n * D + j] + b2[j];
  t = (t - m2[j]) * rsqrtf(vv2[j] + BN_EPS) * g2[j] + bb2[j];
  t = fmaxf(t, 0.0f);
  int g = batch[n];
  atomicAdd(&pooled[g * D + j], t);
  if (j == 0) atomicAdd(&cnt[g], 1.0f);
}

// Classifier layer 1 via WMMA: [64,128] x [128,64], 16 tiles, one block.
__global__ __launch_bounds__(512) void k_cls1(const float* pooled, const float* cnt,
                                              const float* W1, const float* b1,
                                              float* z) {
  const int tid  = threadIdx.x;
  const int wave = tid >> 5;
  const int lane = tid & 31;
  const int half = lane >> 4;
  const int mr   = lane & 15;
  const int rowTile = wave >> 2;
  const int colTile = wave & 3;
  const int rowA    = rowTile * 16 + mr;
  const int colBase = colTile * 16;
  const float rc = 1.0f / fmaxf(cnt[rowA], 1.0f);   // mean-pool division fused in
  v8f acc = {};
#pragma unroll 8
  for (int k0 = 0; k0 < D; k0 += 4) {
    int k = k0 + 2 * half;
    v2f a; a.x = pooled[rowA * D + k] * rc;  a.y = pooled[rowA * D + k + 1] * rc;
    v2f b; b.x = W1[k * 64 + colBase + mr];  b.y = W1[(k + 1) * 64 + colBase + mr];
    acc = __builtin_amdgcn_wmma_f32_16x16x4_f32(false, a, false, b,
                                                (short)0, acc, false, false);
  }
#pragma unroll
  for (int r = 0; r < 8; ++r) {
    int row = rowTile * 16 + r + 8 * half;
    int col = colBase + mr;
    z[row * 64 + col] = fmaxf(acc[r] + b1[col], 0.0f);
  }
}

// Classifier layer 2 + sigmoid: 64 dots of length 64.
__global__ void k_cls2(const float* z, const float* W2, const float* b2,
                       float* out) {
  int g = threadIdx.x;
  float t = b2[0];
  for (int j = 0; j < 64; ++j) t += z[g * 64 + j] * W2[j];
  out[g] = 1.0f / (1.0f + expf(-t));
}

// ---------------------------------------------------------------------------
extern "C" void kernel_launch(void* const* d_in, const int* in_sizes, int n_in,
                              void* d_out, int out_size, void* d_ws, size_t ws_size,
                              hipStream_t stream) {
  const float* x     = (const float*)d_in[0];
  const int*   ei    = (const int*)d_in[1];
  const int*   batch = (const int*)d_in[2];
  const float* encW  = (const float*)d_in[3];
  const float* encb  = (const float*)d_in[4];
  const float* W1    = (const float*)d_in[5];
  const float* b1    = (const float*)d_in[6];
  const float* g1    = (const float*)d_in[7];
  const float* bb1   = (const float*)d_in[8];
  const float* m1    = (const float*)d_in[9];
  const float* vv1   = (const float*)d_in[10];
  const float* W2    = (const float*)d_in[11];
  const float* b2    = (const float*)d_in[12];
  const float* g2    = (const float*)d_in[13];
  const float* bb2   = (const float*)d_in[14];
  const float* m2    = (const float*)d_in[15];
  const float* vv2   = (const float*)d_in[16];
  const float* cW1   = (const float*)d_in[17];
  const float* cb1   = (const float*)d_in[18];
  const float* cW2   = (const float*)d_in[19];
  const float* cb2   = (const float*)d_in[20];
  float* out = (float*)d_out;

  const int N = in_sizes[0];
  const int E = in_sizes[1] / 2;

  auto align256 = [](size_t v) { return (v + 255) & ~(size_t)255; };
  char*  ws  = (char*)d_ws;
  size_t ND  = align256((size_t)N * D * sizeof(float));
  float* h1agg  = (float*)ws;  ws += ND;                 // h1, later aliased as agg
  float* hw2s   = (float*)ws;  ws += ND;
  float* dinv   = (float*)ws;  ws += align256((size_t)N * sizeof(float));
  float* S0     = (float*)ws;  ws += align256((size_t)N * sizeof(float));
  float* S1     = (float*)ws;  ws += align256((size_t)N * sizeof(float));
  float* pooled = (float*)ws;  ws += align256((size_t)G * D * sizeof(float));
  float* cnt    = (float*)ws;  ws += align256((size_t)G * sizeof(float));
  float* vvec   = (float*)ws;  ws += align256((size_t)D * sizeof(float));
  float* cvec   = (float*)ws;  ws += align256((size_t)D * sizeof(float));
  float* zbuf   = (float*)ws;  ws += align256((size_t)G * 64 * sizeof(float));

  const int TB = 256;
  k_init <<<(N + TB - 1) / TB, TB, 0, stream>>>(dinv, S0, S1, pooled, cnt, N);
  k_vc   <<<1, D, 0, stream>>>(encW, encb, W1, vvec, cvec);
  k_deg  <<<(E + TB - 1) / TB, TB, 0, stream>>>(ei, dinv, E);
  k_dinv <<<(N + TB - 1) / TB, TB, 0, stream>>>(dinv, N);
  k_edge1<<<(E + TB - 1) / TB, TB, 0, stream>>>(ei, x, dinv, S0, S1, E);
  k_l1fin<<<N, D, 0, stream>>>(x, dinv, S0, S1, vvec, cvec, b1,
                               g1, bb1, m1, vv1, h1agg, N);
  k_gemm2<<<(N + 15) / 16, 256, 0, stream>>>(h1agg, W2, dinv, hw2s, h1agg, N);
  k_edge2<<<(E + 7) / 8, 256, 0, stream>>>(ei, hw2s, h1agg, E);
  k_fin2 <<<N, D, 0, stream>>>(h1agg, dinv, batch, b2, g2, bb2, m2, vv2,
                               pooled, cnt, N);
  k_cls1 <<<1, 512, 0, stream>>>(pooled, cnt, cW1, cb1, zbuf);
  k_cls2 <<<1, G, 0, stream>>>(zbuf, cW2, cb2, out);
}